// LSTMAction_49194555408734
// MI455X (gfx1250) — compile-verified
//
#include <hip/hip_runtime.h>

// ---------------------------------------------------------------------------
// Sequential LSTM scan (H=16, input_size=1, T=262144) for MI455X / gfx1250.
//
// Latency-bound serial recurrence -> single wave32, whole state in registers.
// Per step, the 64x16 (4 gate blocks of 16x16) W_hh @ h mat-vec runs on the
// matrix pipe as V_WMMA_F32_16X16X4_F32: A = h broadcast across rows (16x4
// per K-chunk), B = W_block^T (4x16 per K-chunk). Each block uses two
// independent depth-2 accumulation chains (only D[0] is consumed, so the
// halves combine with one scalar add). With this mapping, D VGPR0 of block b
// holds gate[b*16 + lane] in lane `lane` (replicated in lanes 16..31), so the
// sigmoid/tanh/c/h update is fully in-lane. A is rebuilt each step from h via
// 8 ds_bpermute_b32. Nonlinearities use gfx1250's native v_tanh_f32 when the
// builtin exists (sigma(x) = 0.5*tanh(x/2)+0.5), else exp2 + v_rcp_f32.
// ---------------------------------------------------------------------------

typedef __attribute__((ext_vector_type(2))) float v2f;
typedef __attribute__((ext_vector_type(8))) float v8f;

#if __has_builtin(__builtin_amdgcn_tanhf)
#define LSTM_HAVE_TANH 1
#else
#define LSTM_HAVE_TANH 0
#endif

__device__ __forceinline__ float fast_exp(float x) {
  return __builtin_amdgcn_exp2f(x * 1.4426950408889634f);  // v_exp_f32
}

__device__ __forceinline__ float tanh_f(float x) {
#if LSTM_HAVE_TANH
  return __builtin_amdgcn_tanhf(x);                        // v_tanh_f32
#else
  float ax = fabsf(x);
  float e  = fast_exp(-2.0f * ax);                         // in (0,1], no inf
  float t  = (1.0f - e) * __builtin_amdgcn_rcpf(1.0f + e); // v_rcp_f32
  return (x < 0.0f) ? -t : t;
#endif
}

__device__ __forceinline__ float sigm(float x) {
#if LSTM_HAVE_TANH
  // sigma(x) = 0.5*tanh(x/2) + 0.5 : mul -> trans -> fma, shortest HW path
  return fmaf(0.5f, __builtin_amdgcn_tanhf(0.5f * x), 0.5f);
#else
  return __builtin_amdgcn_rcpf(1.0f + fast_exp(-x));       // inf-safe
#endif
}

__device__ __forceinline__ float bperm(int idx_byte, float v) {
  return __int_as_float(
      __builtin_amdgcn_ds_bpermute(idx_byte, __float_as_int(v)));
}

__device__ __forceinline__ v8f wmma4(v2f a, v2f b, v8f c) {
  return __builtin_amdgcn_wmma_f32_16x16x4_f32(
      /*neg_a=*/false, a, /*neg_b=*/false, b,
      /*c_mod=*/(short)0, c, /*reuse_a=*/false, /*reuse_b=*/false);
}

// One LSTM timestep. vh/vc hold h[lane&15] / c[lane&15] (replicated in the
// upper lane half). Uniform control flow only: EXEC stays all-ones for WMMA.
__device__ __forceinline__ void lstm_step(float yt, float& vh, float& vc,
                                          const v2f (&Bm)[4][4],
                                          const float (&wih)[4],
                                          const float (&bias)[4],
                                          const int (&iA0)[4],
                                          const int (&iA1)[4]) {
  // Rebuild A (h broadcast) for the 4 K-chunks: 8 lane-gathers from vh.
  v2f A[4];
#pragma unroll
  for (int c = 0; c < 4; ++c) {
    A[c].x = bperm(iA0[c], vh);
    A[c].y = bperm(iA1[c], vh);
  }

  // Per gate block: two independent depth-2 chains (chunks {0,1} and {2,3}),
  // interleaved across blocks so the matrix pipe never sees a short-distance
  // D->C dependency. Only element [0] of each D is consumed.
  v8f dA[4], dB[4];
#pragma unroll
  for (int b = 0; b < 4; ++b) dA[b] = wmma4(A[0], Bm[b][0], (v8f){});
#pragma unroll
  for (int b = 0; b < 4; ++b) dB[b] = wmma4(A[2], Bm[b][2], (v8f){});
#pragma unroll
  for (int b = 0; b < 4; ++b) dA[b] = wmma4(A[1], Bm[b][1], dA[b]);
#pragma unroll
  for (int b = 0; b < 4; ++b) dB[b] = wmma4(A[3], Bm[b][3], dB[b]);

  // gates[b*16 + lane] = (dA+dB)[0] + (y[t]*W_ih + b_ih + b_hh)
  float gi = (dA[0][0] + dB[0][0]) + fmaf(yt, wih[0], bias[0]);
  float gf = (dA[1][0] + dB[1][0]) + fmaf(yt, wih[1], bias[1]);
  float gg = (dA[2][0] + dB[2][0]) + fmaf(yt, wih[2], bias[2]);
  float go = (dA[3][0] + dB[3][0]) + fmaf(yt, wih[3], bias[3]);

  float i_ = sigm(gi);
  float f_ = sigm(gf);
  float g_ = tanh_f(gg);
  float o_ = sigm(go);

  vc = fmaf(f_, vc, i_ * g_);
  vh = o_ * tanh_f(vc);
}

__global__ void __launch_bounds__(32, 1)
lstm_scan_wmma(const float* __restrict__ y, const float* __restrict__ x_prev,
               const float* __restrict__ W_ih, const float* __restrict__ W_hh,
               const float* __restrict__ b_ih, const float* __restrict__ b_hh,
               const float* __restrict__ fc_w, const float* __restrict__ fc_b,
               float* __restrict__ out, int T) {
  const int lane = threadIdx.x;   // wave32: 0..31
  const int n    = lane & 15;     // column / hidden index
  const int half = lane >> 4;     // 0: K=0,1  1: K=2,3 (WMMA 16x16x4 layout)

  // ---- One-time setup: B = W_block^T in WMMA B-operand layout ------------
  // B VGPR0 holds K = 2*half, VGPR1 holds K = 2*half+1; B[k][n] = W[b*16+n][4c+k].
  v2f Bm[4][4];
#pragma unroll
  for (int b = 0; b < 4; ++b) {
#pragma unroll
    for (int c = 0; c < 4; ++c) {
      const int row = b * 16 + n;           // gate index within W_hh [64][16]
      const int k0  = 4 * c + 2 * half;
      Bm[b][c].x = W_hh[row * 16 + k0];
      Bm[b][c].y = W_hh[row * 16 + k0 + 1];
    }
  }

  // Input-side projection constants: x_proj[t][g] = y[t]*W_ih[g] + b_ih[g]+b_hh[g]
  float wih[4], bias[4];
#pragma unroll
  for (int b = 0; b < 4; ++b) {
    const int g = b * 16 + n;
    wih[b]  = W_ih[g];                      // W_ih is [64][1]
    bias[b] = b_ih[g] + b_hh[g];
  }

  // Constant bpermute byte-indices to rebuild A[m][k] = h[k] each step:
  // A chunk c: VGPR0 needs h[4c + 2*half], VGPR1 needs h[4c + 2*half + 1].
  int iA0[4], iA1[4];
#pragma unroll
  for (int c = 0; c < 4; ++c) {
    iA0[c] = (4 * c + 2 * half) << 2;
    iA1[c] = iA0[c] + 4;
  }

  // ---- The scan -----------------------------------------------------------
  float vh = 0.0f;   // h[lane&15]
  float vc = 0.0f;   // c[lane&15]

  const int Tmain = T & ~7;
  for (int t0 = 0; t0 < Tmain; t0 += 8) {
    // Uniform addresses -> one s_load_b256, batched ahead of 8 serial steps.
    float ybuf[8];
#pragma unroll
    for (int u = 0; u < 8; ++u) ybuf[u] = y[t0 + u];
#pragma unroll
    for (int u = 0; u < 8; ++u)
      lstm_step(ybuf[u], vh, vc, Bm, wih, bias, iA0, iA1);
  }
  for (int t = Tmain; t < T; ++t)           // uniform remainder (T%8 != 0 case)
    lstm_step(y[t], vh, vc, Bm, wih, bias, iA0, iA1);

  // ---- Final FC: fc_w @ [h ; x_prev] + fc_b ------------------------------
  float part = (lane < 16) ? vh * fc_w[lane] : 0.0f;
#pragma unroll
  for (int off = 16; off > 0; off >>= 1)
    part += __shfl_down(part, off);
  if (lane == 0)
    out[0] = part + fc_w[16] * x_prev[0] + fc_b[0];
}

extern "C" void kernel_launch(void* const* d_in, const int* in_sizes, int n_in,
                              void* d_out, int out_size, void* d_ws,
                              size_t ws_size, hipStream_t stream) {
  const float* y      = (const float*)d_in[0];
  const float* x_prev = (const float*)d_in[1];
  const float* W_ih   = (const float*)d_in[2];
  const float* W_hh   = (const float*)d_in[3];
  const float* b_ih   = (const float*)d_in[4];
  const float* b_hh   = (const float*)d_in[5];
  const float* fc_w   = (const float*)d_in[6];
  const float* fc_b   = (const float*)d_in[7];
  const int T = in_sizes[0];

  // Serial recurrence: exactly one wave32, everything register-resident.
  lstm_scan_wmma<<<1, 32, 0, stream>>>(y, x_prev, W_ih, W_hh, b_ih, b_hh,
                                       fc_w, fc_b, (float*)d_out, T);
}